// SBoRAFAModule_6390911336859
// MI455X (gfx1250) — compile-verified
//
#include <hip/hip_runtime.h>

typedef __attribute__((ext_vector_type(2))) float v2f;
typedef __attribute__((ext_vector_type(8))) float v8f;

#define D_IN     4096
#define D_OUT    4096
#define LORA_DIM 64
#define M_TOTAL  (4 * 4096)

constexpr int BM  = 128;  // rows per block (8 waves x 16)
constexpr int BN  = 64;   // cols per block (4 wmma tiles per wave)
constexpr int LDA = 66;   // padded LDS stride (keeps 8B align, spreads banks)
constexpr int LDB = 66;

__launch_bounds__(256)
__global__ void sbora_lora_gemm(const float* __restrict__ x,
                                const float* __restrict__ w,
                                const int*   __restrict__ idx,
                                float*       __restrict__ out) {
  __shared__ float sA[BM * LDA];  // gathered x tile: sA[m][k]
  __shared__ float sB[BN * LDB];  // weight tile:     sB[n][k]

  const int tid   = threadIdx.x;
  const int mBase = blockIdx.y * BM;
  const int nBase = blockIdx.x * BN;

  // ---- cooperative gather: sA[m][k] = x[(mBase+m)*D_IN + idx[k]] ----------
  {
    const int k   = tid & 63;       // each thread owns one gathered column
    const int mo  = tid >> 6;       // 0..3
    const int col = idx[k];
    const float* xp = x + (size_t)(mBase + mo) * D_IN + col;
    #pragma unroll
    for (int i = 0; i < 32; ++i) {  // m = mo + 4*i covers 0..127
      sA[(mo + 4 * i) * LDA + k] = xp[(size_t)(4 * i) * D_IN];
    }
  }
  // ---- load B tile (coalesced): sB[n][k] = w[(nBase+n)*LORA_DIM + k] ------
  {
    #pragma unroll
    for (int i = 0; i < 16; ++i) {
      const int flat = i * 256 + tid;
      const int n = flat >> 6;
      const int k = flat & 63;
      sB[n * LDB + k] = w[(size_t)(nBase + n) * LORA_DIM + k];
    }
  }
  __syncthreads();

  const int wave = tid >> 5;
  const int lane = tid & 31;
  const int half = lane >> 4;  // 0: lanes 0-15, 1: lanes 16-31
  const int l16  = lane & 15;

  v8f acc0 = {}, acc1 = {}, acc2 = {}, acc3 = {};

  const float* aRow = &sA[(wave * 16 + l16) * LDA];
  const float* bR0  = &sB[(0 * 16 + l16) * LDB];
  const float* bR1  = &sB[(1 * 16 + l16) * LDB];
  const float* bR2  = &sB[(2 * 16 + l16) * LDB];
  const float* bR3  = &sB[(3 * 16 + l16) * LDB];

  // K = 64 in 16 steps of 16x16x4 fp32 WMMA; A/B VGPR r <-> K = k0 + 2*half + r
  #pragma unroll
  for (int k0 = 0; k0 < LORA_DIM; k0 += 4) {
    const int kk = k0 + 2 * half;
    v2f a;
    a.x = aRow[kk];
    a.y = aRow[kk + 1];
    v2f b;
    b.x = bR0[kk]; b.y = bR0[kk + 1];
    acc0 = __builtin_amdgcn_wmma_f32_16x16x4_f32(false, a, false, b, (short)0, acc0, false, false);
    b.x = bR1[kk]; b.y = bR1[kk + 1];
    acc1 = __builtin_amdgcn_wmma_f32_16x16x4_f32(false, a, false, b, (short)0, acc1, false, false);
    b.x = bR2[kk]; b.y = bR2[kk + 1];
    acc2 = __builtin_amdgcn_wmma_f32_16x16x4_f32(false, a, false, b, (short)0, acc2, false, false);
    b.x = bR3[kk]; b.y = bR3[kk + 1];
    acc3 = __builtin_amdgcn_wmma_f32_16x16x4_f32(false, a, false, b, (short)0, acc3, false, false);
  }

  // ---- store: C VGPR r holds row (r + 8*half), col = l16 within each tile -
  constexpr float OUT_SCALE = 1.0f;  // (ALPHA / LORA_DIM) * SCALAR
  float* op = out + (size_t)(mBase + wave * 16 + 8 * half) * D_OUT + nBase + l16;
  #pragma unroll
  for (int r = 0; r < 8; ++r) {
    float* orow = op + (size_t)r * D_OUT;
    orow[0]  = acc0[r] * OUT_SCALE;
    orow[16] = acc1[r] * OUT_SCALE;
    orow[32] = acc2[r] * OUT_SCALE;
    orow[48] = acc3[r] * OUT_SCALE;
  }
}

extern "C" void kernel_launch(void* const* d_in, const int* in_sizes, int n_in,
                              void* d_out, int out_size, void* d_ws, size_t ws_size,
                              hipStream_t stream) {
  const float* x   = (const float*)d_in[0];  // [4, 4096, 4096] fp32
  const float* w   = (const float*)d_in[1];  // [4096, 64] fp32
  const int*   idx = (const int*)d_in[2];    // [64] indices
  float*       out = (float*)d_out;          // [4, 4096, 4096] fp32

  dim3 grid(D_OUT / BN, M_TOTAL / BM, 1);    // (64, 128): N-major for L2 reuse of A
  sbora_lora_gemm<<<grid, dim3(256), 0, stream>>>(x, w, idx, out);
}